// MiMoV2MoE_2044404433735
// MI455X (gfx1250) — compile-verified
//
#include <hip/hip_runtime.h>
#include <hip/hip_bf16.h>
#include <math.h>

typedef float v2f __attribute__((ext_vector_type(2)));
typedef float v8f __attribute__((ext_vector_type(8)));
typedef int v4i __attribute__((vector_size(16)));
typedef __attribute__((address_space(1))) v4i gv4i;  // global
typedef __attribute__((address_space(3))) v4i lv4i;  // LDS

#if __has_builtin(__builtin_amdgcn_global_load_async_to_lds_b128)
#define MOE_ASYNC 1
#else
#define MOE_ASYNC 0
#endif

#if MOE_ASYNC && __has_builtin(__builtin_amdgcn_s_wait_asynccnt)
#define MOE_WAIT_ASYNC(n) __builtin_amdgcn_s_wait_asynccnt(n)
#else
#define MOE_WAIT_ASYNC(n)
#endif

namespace {

constexpr int kT = 2048;      // tokens
constexpr int kH = 1024;      // hidden
constexpr int kI = 512;       // moe intermediate
constexpr int kE = 64;        // experts
constexpr int kNG = 8;        // router groups
constexpr int kTopG = 4;      // top groups
constexpr int kTopK = 6;      // experts per token
constexpr int kRows = kT * kTopK;  // 12288 dispatched rows

// X tiles: rows of 36 floats (144B: 16B aligned for async B128, conflict-free
// 8B fragment loads).  W tiles: rows of 72 floats (288B: 16B aligned, and
// 2*72 mod 64 = 16 keeps the two half-wave K-rows on disjoint LDS banks).
constexpr int kXP = 36;
constexpr int kWP = 72;

__device__ __forceinline__ v8f wmma_f32_k4(v2f a, v2f b, v8f c) {
  // V_WMMA_F32_16X16X4_F32 : D = A(16x4) * B(4x16) + C(16x16), full fp32
  return __builtin_amdgcn_wmma_f32_16x16x4_f32(false, a, false, b, (short)0, c,
                                               false, false);
}

// 16B global -> LDS copy; DMA (ASYNCcnt) when the toolchain exposes it.
__device__ __forceinline__ void cp16(const float* g, float* l) {
#if MOE_ASYNC
  __builtin_amdgcn_global_load_async_to_lds_b128((gv4i*)(g), (lv4i*)(l), 0, 0);
#else
  *(float4*)l = *(const float4*)g;
#endif
}

// ---------------------------------------------------------------- meta zero
__global__ void zero_meta_kernel(int* __restrict__ counts, int* __restrict__ cursor) {
  int i = threadIdx.x;
  if (i < kE) { counts[i] = 0; cursor[i] = 0; }
}

// ---------------------------------------------------------------- router GEMM
__global__ __launch_bounds__(128) void router_kernel(
    const float* __restrict__ hs, const float* __restrict__ gw,
    float* __restrict__ scores) {
  const int t0 = blockIdx.x * 64;
  const int tid = threadIdx.x;
  const int lane = tid & 31, wave = tid >> 5;
  const int wm = wave >> 1, wn = wave & 1;
  const int half = lane >> 4, l16 = lane & 15;

  __shared__ float Xs[64][34];
  __shared__ float Gt[64][34];

  v8f acc[2][2];
#pragma unroll
  for (int i = 0; i < 2; ++i)
#pragma unroll
    for (int j = 0; j < 2; ++j) acc[i][j] = {};

  for (int k0 = 0; k0 < kH; k0 += 32) {
#pragma unroll
    for (int i = 0; i < 16; ++i) {
      int idx = i * 128 + tid;
      int m = idx >> 5, kk = idx & 31;
      Xs[m][kk] = hs[(size_t)(t0 + m) * kH + k0 + kk];
      Gt[m][kk] = gw[(size_t)m * kH + k0 + kk];  // m doubles as expert index
    }
    __syncthreads();
#pragma unroll
    for (int kk = 0; kk < 32; kk += 4) {
      const int ko = kk + half * 2;
      v2f a0 = *(const v2f*)&Xs[wm * 32 + l16][ko];
      v2f a1 = *(const v2f*)&Xs[wm * 32 + 16 + l16][ko];
      v2f b0 = *(const v2f*)&Gt[wn * 32 + l16][ko];
      v2f b1 = *(const v2f*)&Gt[wn * 32 + 16 + l16][ko];
      acc[0][0] = wmma_f32_k4(a0, b0, acc[0][0]);
      acc[0][1] = wmma_f32_k4(a0, b1, acc[0][1]);
      acc[1][0] = wmma_f32_k4(a1, b0, acc[1][0]);
      acc[1][1] = wmma_f32_k4(a1, b1, acc[1][1]);
    }
    __syncthreads();
  }
#pragma unroll
  for (int i = 0; i < 2; ++i)
#pragma unroll
    for (int j = 0; j < 2; ++j) {
      v8f c = acc[i][j];
#pragma unroll
      for (int v = 0; v < 8; ++v) {
        int t = t0 + wm * 32 + i * 16 + half * 8 + v;
        int e = wn * 32 + j * 16 + l16;
        scores[(size_t)t * kE + e] = 1.0f / (1.0f + __expf(-c[v]));
      }
    }
}

// ---------------------------------------------------------------- grouped top-k
__global__ __launch_bounds__(256) void topk_kernel(
    const float* __restrict__ scores, const float* __restrict__ ebias,
    int* __restrict__ topk_ids, float* __restrict__ topk_w,
    int* __restrict__ counts) {
  const int t = blockIdx.x * blockDim.x + threadIdx.x;
  if (t >= kT) return;

  float sfc[kE];
#pragma unroll
  for (int e = 0; e < kE; ++e) sfc[e] = scores[(size_t)t * kE + e] + ebias[e];

  float gs[kNG];
#pragma unroll
  for (int g = 0; g < kNG; ++g) {
    float m1 = -INFINITY, m2 = -INFINITY;
#pragma unroll
    for (int j = 0; j < kE / kNG; ++j) {
      float v = sfc[g * (kE / kNG) + j];
      if (v > m1) { m2 = m1; m1 = v; } else if (v > m2) { m2 = v; }
    }
    gs[g] = m1 + m2;
  }
  unsigned gsel = 0;
#pragma unroll
  for (int it = 0; it < kTopG; ++it) {
    float b = -INFINITY; int bi = 0;
#pragma unroll
    for (int g = 0; g < kNG; ++g)
      if (!((gsel >> g) & 1) && gs[g] > b) { b = gs[g]; bi = g; }
    gsel |= 1u << bi;
  }
  unsigned long long esel = 0;
  int ids[kTopK]; float wv[kTopK]; float wsum = 0.0f;
#pragma unroll
  for (int it = 0; it < kTopK; ++it) {
    float b = -INFINITY; int bi = 0;
#pragma unroll
    for (int e = 0; e < kE; ++e) {
      if (((gsel >> (e / (kE / kNG))) & 1) && !((esel >> e) & 1) && sfc[e] > b) {
        b = sfc[e]; bi = e;
      }
    }
    esel |= 1ull << bi;
    ids[it] = bi;
    wv[it] = b - ebias[bi];  // unbiased sigmoid score used as weight
    wsum += wv[it];
  }
  const float inv = 1.0f / wsum;
#pragma unroll
  for (int k = 0; k < kTopK; ++k) {
    topk_ids[t * kTopK + k] = ids[k];
    topk_w[t * kTopK + k] = wv[k] * inv;
    atomicAdd(&counts[ids[k]], 1);
  }
}

// ---------------------------------------------------------------- scan + assign
__global__ void scan_kernel(const int* __restrict__ counts, int* __restrict__ offs) {
  if (threadIdx.x == 0) {
    int s = 0;
    for (int e = 0; e < kE; ++e) { offs[e] = s; s += counts[e]; }
    offs[kE] = s;
  }
}

__global__ __launch_bounds__(256) void assign_kernel(
    const int* __restrict__ topk_ids, const int* __restrict__ offs,
    int* __restrict__ cursor, int* __restrict__ row_token,
    int* __restrict__ row_slot) {
  const int s = blockIdx.x * blockDim.x + threadIdx.x;
  if (s >= kRows) return;
  const int e = topk_ids[s];
  const int r = offs[e] + atomicAdd(&cursor[e], 1);
  row_token[r] = s / kTopK;
  row_slot[r] = s;
}

// ---------------------------------------------------------------- up proj: SwiGLU
// act[r, n] = silu(x@w1)[r,n] * (x@w3)[r,n], async double-buffered staging.
__global__ __launch_bounds__(128) void moe_up_kernel(
    const float* __restrict__ hs, const float* __restrict__ w1,
    const float* __restrict__ w3, const int* __restrict__ offs,
    const int* __restrict__ row_token, float* __restrict__ act) {
  const int e = blockIdx.z, mt = blockIdx.y, nt = blockIdx.x;
  const int r0 = offs[e], r1 = offs[e + 1];
  const int rb = r0 + mt * 64;
  if (rb >= r1) return;

  const int tid = threadIdx.x;
  const int lane = tid & 31, wave = tid >> 5;
  const int wm = wave >> 1, wn = wave & 1;
  const int half = lane >> 4, l16 = lane & 15;

  __shared__ float Xs[2][64][kXP];
  __shared__ float W1s[2][32][kWP];
  __shared__ float W3s[2][32][kWP];
  __shared__ int tok[64];

  if (tid < 64) {
    int r = rb + tid;
    tok[tid] = (r < r1) ? row_token[r] : 0;  // pad rows: any valid token (discarded)
  }
  __syncthreads();

  // per-thread staging coordinates (fixed across chunks)
  const int xq = (tid & 7) * 4;    // float4 column within X row
  const int xm0 = tid >> 3;        // X row, +16 per iteration
  const int wq = (tid & 15) * 4;   // float4 column within W row
  const int wk0 = tid >> 4;        // W k-row, +8 per iteration

  const float* w1base = w1 + (size_t)e * kH * kI + nt * 64;
  const float* w3base = w3 + (size_t)e * kH * kI + nt * 64;

  auto stage = [&](int b, int k0) {
#pragma unroll
    for (int i = 0; i < 4; ++i) {
      int m = xm0 + i * 16;
      cp16(hs + (size_t)tok[m] * kH + k0 + xq, &Xs[b][m][xq]);
    }
#pragma unroll
    for (int i = 0; i < 4; ++i) {
      int kk = wk0 + i * 8;
      cp16(w1base + (size_t)(k0 + kk) * kI + wq, &W1s[b][kk][wq]);
      cp16(w3base + (size_t)(k0 + kk) * kI + wq, &W3s[b][kk][wq]);
    }
  };

  v8f accg[2][2], accu[2][2];
#pragma unroll
  for (int i = 0; i < 2; ++i)
#pragma unroll
    for (int j = 0; j < 2; ++j) { accg[i][j] = {}; accu[i][j] = {}; }

  const int n0 = wn * 32 + l16, n1 = n0 + 16;
  const int m0r = wm * 32 + l16, m1r = m0r + 16;

  stage(0, 0);  // 12 async ops per wave in flight
  constexpr int NK = kH / 32;
  for (int c = 0; c < NK; ++c) {
    const int b = c & 1;
    if (c + 1 < NK) {
      stage(b ^ 1, (c + 1) * 32);  // prefetch next chunk (12 more in flight)
      MOE_WAIT_ASYNC(12);          // chunk c landed; c+1 may still fly
    } else {
      MOE_WAIT_ASYNC(0);
    }
    __syncthreads();
#pragma unroll
    for (int kk = 0; kk < 32; kk += 4) {
      const int ko = kk + half * 2;
      v2f a0 = *(const v2f*)&Xs[b][m0r][ko];
      v2f a1 = *(const v2f*)&Xs[b][m1r][ko];
      v2f g0, g1, u0, u1;
      g0.x = W1s[b][ko][n0]; g0.y = W1s[b][ko + 1][n0];
      g1.x = W1s[b][ko][n1]; g1.y = W1s[b][ko + 1][n1];
      u0.x = W3s[b][ko][n0]; u0.y = W3s[b][ko + 1][n0];
      u1.x = W3s[b][ko][n1]; u1.y = W3s[b][ko + 1][n1];
      accg[0][0] = wmma_f32_k4(a0, g0, accg[0][0]);
      accg[0][1] = wmma_f32_k4(a0, g1, accg[0][1]);
      accg[1][0] = wmma_f32_k4(a1, g0, accg[1][0]);
      accg[1][1] = wmma_f32_k4(a1, g1, accg[1][1]);
      accu[0][0] = wmma_f32_k4(a0, u0, accu[0][0]);
      accu[0][1] = wmma_f32_k4(a0, u1, accu[0][1]);
      accu[1][0] = wmma_f32_k4(a1, u0, accu[1][0]);
      accu[1][1] = wmma_f32_k4(a1, u1, accu[1][1]);
    }
    __syncthreads();  // everyone done with buf b before it is restaged
  }

  const int mrem = r1 - rb;
#pragma unroll
  for (int i = 0; i < 2; ++i)
#pragma unroll
    for (int j = 0; j < 2; ++j) {
      v8f g = accg[i][j], u = accu[i][j];
#pragma unroll
      for (int v = 0; v < 8; ++v) {
        int ml = wm * 32 + i * 16 + half * 8 + v;
        if (ml < mrem) {
          float gv = g[v];
          float s = gv / (1.0f + __expf(-gv));  // silu
          int n = nt * 64 + wn * 32 + j * 16 + l16;
          act[(size_t)(rb + ml) * kI + n] = s * u[v];
        }
      }
    }
}

// ---------------------------------------------------------------- down proj
__global__ __launch_bounds__(128) void moe_down_kernel(
    const float* __restrict__ act, const float* __restrict__ w2,
    const int* __restrict__ offs, const int* __restrict__ row_slot,
    float* __restrict__ slot_out) {
  const int e = blockIdx.z, mt = blockIdx.y, nt = blockIdx.x;
  const int r0 = offs[e], r1 = offs[e + 1];
  const int rb = r0 + mt * 64;
  if (rb >= r1) return;

  const int tid = threadIdx.x;
  const int lane = tid & 31, wave = tid >> 5;
  const int wm = wave >> 1, wn = wave & 1;
  const int half = lane >> 4, l16 = lane & 15;

  __shared__ float As[2][64][kXP];
  __shared__ float Ws[2][32][kWP];
  __shared__ int slt[64];

  if (tid < 64) {
    int r = rb + tid;
    slt[tid] = (r < r1) ? row_slot[r] : 0;
  }

  const int xq = (tid & 7) * 4;
  const int xm0 = tid >> 3;
  const int wq = (tid & 15) * 4;
  const int wk0 = tid >> 4;

  const float* wbase = w2 + (size_t)e * kI * kH + nt * 64;

  auto stage = [&](int b, int k0) {
#pragma unroll
    for (int i = 0; i < 4; ++i) {
      int m = xm0 + i * 16;
      cp16(act + (size_t)(rb + m) * kI + k0 + xq, &As[b][m][xq]);
    }
#pragma unroll
    for (int i = 0; i < 4; ++i) {
      int kk = wk0 + i * 8;
      cp16(wbase + (size_t)(k0 + kk) * kH + wq, &Ws[b][kk][wq]);
    }
  };

  v8f acc[2][2];
#pragma unroll
  for (int i = 0; i < 2; ++i)
#pragma unroll
    for (int j = 0; j < 2; ++j) acc[i][j] = {};

  const int n0 = wn * 32 + l16, n1 = n0 + 16;
  const int m0r = wm * 32 + l16, m1r = m0r + 16;

  stage(0, 0);  // 8 async ops per wave in flight
  constexpr int NK = kI / 32;
  for (int c = 0; c < NK; ++c) {
    const int b = c & 1;
    if (c + 1 < NK) {
      stage(b ^ 1, (c + 1) * 32);
      MOE_WAIT_ASYNC(8);
    } else {
      MOE_WAIT_ASYNC(0);
    }
    __syncthreads();
#pragma unroll
    for (int kk = 0; kk < 32; kk += 4) {
      const int ko = kk + half * 2;
      v2f a0 = *(const v2f*)&As[b][m0r][ko];
      v2f a1 = *(const v2f*)&As[b][m1r][ko];
      v2f b0, b1;
      b0.x = Ws[b][ko][n0]; b0.y = Ws[b][ko + 1][n0];
      b1.x = Ws[b][ko][n1]; b1.y = Ws[b][ko + 1][n1];
      acc[0][0] = wmma_f32_k4(a0, b0, acc[0][0]);
      acc[0][1] = wmma_f32_k4(a0, b1, acc[0][1]);
      acc[1][0] = wmma_f32_k4(a1, b0, acc[1][0]);
      acc[1][1] = wmma_f32_k4(a1, b1, acc[1][1]);
    }
    __syncthreads();
  }

  const int mrem = r1 - rb;
#pragma unroll
  for (int i = 0; i < 2; ++i)
#pragma unroll
    for (int j = 0; j < 2; ++j) {
      v8f c = acc[i][j];
#pragma unroll
      for (int v = 0; v < 8; ++v) {
        int ml = wm * 32 + i * 16 + half * 8 + v;
        if (ml < mrem) {
          int s = slt[ml];
          int n = nt * 64 + wn * 32 + j * 16 + l16;
          slot_out[(size_t)s * kH + n] = c[v];
        }
      }
    }
}

// ---------------------------------------------------------------- combine
__global__ __launch_bounds__(256) void combine_kernel(
    const float* __restrict__ slot_out, const float* __restrict__ topk_w,
    float* __restrict__ out) {
  const int t = blockIdx.x;
  float w[kTopK];
#pragma unroll
  for (int k = 0; k < kTopK; ++k) w[k] = topk_w[t * kTopK + k];
  const float* base = slot_out + (size_t)t * kTopK * kH;
  for (int h = threadIdx.x; h < kH; h += 256) {
    float acc = 0.0f;
#pragma unroll
    for (int k = 0; k < kTopK; ++k) acc += w[k] * base[(size_t)k * kH + h];
    out[(size_t)t * kH + h] = acc;
  }
}

}  // namespace

extern "C" void kernel_launch(void* const* d_in, const int* in_sizes, int n_in,
                              void* d_out, int out_size, void* d_ws, size_t ws_size,
                              hipStream_t stream) {
  (void)in_sizes; (void)n_in; (void)out_size; (void)ws_size;
  const float* hs = (const float*)d_in[0];   // [T,H]
  const float* gw = (const float*)d_in[1];   // [E,H]
  const float* eb = (const float*)d_in[2];   // [E]
  const float* w1 = (const float*)d_in[3];   // [E,H,I]
  const float* w3 = (const float*)d_in[4];   // [E,H,I]
  const float* w2 = (const float*)d_in[5];   // [E,I,H]
  float* out = (float*)d_out;                // [T,H]

  char* p = (char*)d_ws;
  auto carve = [&](size_t bytes) {
    char* r = p;
    p += (bytes + 255) & ~size_t(255);
    return r;
  };
  float* scores    = (float*)carve((size_t)kT * kE * 4);
  int*   tids      = (int*)carve((size_t)kRows * 4);
  float* tw        = (float*)carve((size_t)kRows * 4);
  int*   counts    = (int*)carve(kE * 4);
  int*   cursor    = (int*)carve(kE * 4);
  int*   offs      = (int*)carve((kE + 1) * 4);
  int*   row_token = (int*)carve((size_t)kRows * 4);
  int*   row_slot  = (int*)carve((size_t)kRows * 4);
  float* act       = (float*)carve((size_t)kRows * kI * 4);
  float* slot_out  = (float*)carve((size_t)kRows * kH * 4);

  zero_meta_kernel<<<1, 128, 0, stream>>>(counts, cursor);
  router_kernel<<<dim3(kT / 64), 128, 0, stream>>>(hs, gw, scores);
  topk_kernel<<<dim3(kT / 256), 256, 0, stream>>>(scores, eb, tids, tw, counts);
  scan_kernel<<<1, 32, 0, stream>>>(counts, offs);
  assign_kernel<<<dim3((kRows + 255) / 256), 256, 0, stream>>>(tids, offs, cursor,
                                                               row_token, row_slot);
  moe_up_kernel<<<dim3(kI / 64, kT / 64, kE), 128, 0, stream>>>(hs, w1, w3, offs,
                                                                row_token, act);
  moe_down_kernel<<<dim3(kH / 64, kT / 64, kE), 128, 0, stream>>>(act, w2, offs,
                                                                  row_slot, slot_out);
  combine_kernel<<<kT, 256, 0, stream>>>(slot_out, tw, out);
}